// SpatialTransformer_63806034149460
// MI455X (gfx1250) — compile-verified
//
#include <hip/hip_runtime.h>
#include <cstddef>

// Problem constants (SIZE = (D,H,W) = (160,192,160), B=2, C=2, fp32)
#define DD 160
#define HH 192
#define WW 160
#define NN (DD * HH * WW)      // 4,915,200 spatial voxels
#define BB 2
#define CC 2

// Reproduce the reference's coordinate round-trip:
//   t  = 2*((g+f)/(dim-1) - 0.5)
//   px = (t+1)*0.5*(dim-1)
__device__ __forceinline__ float ref_px(float g, float f, float dm1) {
    float t = 2.0f * ((g + f) / dm1 - 0.5f);
    return (t + 1.0f) * 0.5f * dm1;
}

__global__ void __launch_bounds__(256)
warp3d_trilinear_kernel(const float* __restrict__ src,
                        const float* __restrict__ flow,
                        float* __restrict__ out)
{
    // batch lives in blockIdx.y -> all batch-dependent bases are scalar (SALU)
    const int b = (int)blockIdx.y;
    const float* __restrict__ flb = flow + (size_t)b * (size_t)(3 * NN);
    const float* __restrict__ s0  = src  + (size_t)b * (size_t)(CC * NN);  // channel 0
    float* __restrict__       ob  = out  + (size_t)b * (size_t)(CC * NN);

    const int stride = (int)(gridDim.x * blockDim.x);

    for (int i = (int)(blockIdx.x * blockDim.x + threadIdx.x); i < NN; i += stride) {
        // ---- CDNA5 prefetch path: pull next iteration's flow triple toward us
        int inext = i + stride;
        if (inext < NN) {
            __builtin_prefetch(flb + inext, 0, 0);                     // global_prefetch_b8
            __builtin_prefetch(flb + inext + (size_t)NN, 0, 0);
            __builtin_prefetch(flb + inext + (size_t)(2 * NN), 0, 0);
        }

        // ---- decode (z, y, x); x fastest -> coalesced within wave32
        int x  = i % WW;
        int t2 = i / WW;
        int y  = t2 % HH;
        int z  = t2 / HH;

        // ---- flow components: 3 coalesced streams, read-once -> non-temporal
        float flx = __builtin_nontemporal_load(flb + i);
        float fly = __builtin_nontemporal_load(flb + i + (size_t)NN);
        float flz = __builtin_nontemporal_load(flb + i + (size_t)(2 * NN));

        // ---- sample position (matches reference FP path)
        float px = ref_px((float)x, flx, (float)(WW - 1));
        float py = ref_px((float)y, fly, (float)(HH - 1));
        float pz = ref_px((float)z, flz, (float)(DD - 1));

        float xf = floorf(px), yf = floorf(py), zf = floorf(pz);
        int   x0 = (int)xf,    y0 = (int)yf,    z0 = (int)zf;
        float fx = px - xf,    fy = py - yf,    fz = pz - zf;

        // per-axis weights and clamped indices / validity
        float wx[2] = {1.0f - fx, fx};
        float wy[2] = {1.0f - fy, fy};
        float wz[2] = {1.0f - fz, fz};

        int xs[2] = {x0, x0 + 1};
        int ys[2] = {y0, y0 + 1};
        int zs[2] = {z0, z0 + 1};

        bool vx[2], vy[2], vz[2];
        int  xc[2], yc[2], zc[2];
#pragma unroll
        for (int k = 0; k < 2; ++k) {
            vx[k] = (xs[k] >= 0) && (xs[k] < WW);
            vy[k] = (ys[k] >= 0) && (ys[k] < HH);
            vz[k] = (zs[k] >= 0) && (zs[k] < DD);
            xc[k] = min(max(xs[k], 0), WW - 1);
            yc[k] = min(max(ys[k], 0), HH - 1);
            zc[k] = min(max(zs[k], 0), DD - 1);
        }

        // precompute plane/row offsets
        int zoff[2] = { zc[0] * (HH * WW), zc[1] * (HH * WW) };
        int yoff[2] = { yc[0] * WW,        yc[1] * WW        };

        float acc0 = 0.0f;
        float acc1 = 0.0f;

        // Reference accumulation order: dz outer, dy, dx inner;
        // weight association ((wx*wy)*wz); invalid corner -> weight 0, clamped gather.
        // src gathers stay regular-temporal: heavy cross-thread cacheline reuse in L2.
#pragma unroll
        for (int dz = 0; dz < 2; ++dz) {
#pragma unroll
            for (int dy = 0; dy < 2; ++dy) {
#pragma unroll
                for (int dx = 0; dx < 2; ++dx) {
                    bool  valid = vx[dx] && vy[dy] && vz[dz];
                    float w     = (wx[dx] * wy[dy]) * wz[dz];
                    w = valid ? w : 0.0f;
                    int off = zoff[dz] + yoff[dy] + xc[dx];
                    float v0 = s0[off];                 // channel 0 gather
                    float v1 = s0[off + NN];            // channel 1 gather
                    acc0 += w * v0;
                    acc1 += w * v1;
                }
            }
        }

        // write-once outputs -> non-temporal stores (don't evict src from L2)
        __builtin_nontemporal_store(acc0, ob + i);
        __builtin_nontemporal_store(acc1, ob + i + (size_t)NN);
    }
}

extern "C" void kernel_launch(void* const* d_in, const int* in_sizes, int n_in,
                              void* d_out, int out_size, void* d_ws, size_t ws_size,
                              hipStream_t stream) {
    (void)in_sizes; (void)n_in; (void)out_size; (void)d_ws; (void)ws_size;
    const float* src  = (const float*)d_in[0];  // [2,2,160,192,160] fp32
    const float* flow = (const float*)d_in[1];  // [2,3,160,192,160] fp32
    float* out        = (float*)d_out;          // [2,2,160,192,160] fp32

    dim3 block(256);
    dim3 grid(2048, BB);   // grid-stride over spatial; batch in y (scalar bases)
    warp3d_trilinear_kernel<<<grid, block, 0, stream>>>(src, flow, out);
}